// TwoRealTwoImaginaryGCNLayer_31293131719204
// MI455X (gfx1250) — compile-verified
//
#include <hip/hip_runtime.h>
#include <math.h>

typedef __attribute__((ext_vector_type(2))) float v2f;
typedef __attribute__((ext_vector_type(8))) float v8f;

#define D 128  // feature dim (in == out)

// ---------------- degree / normalization ----------------

__global__ void deg_init_kernel(float* deg, int n) {
    int i = blockIdx.x * blockDim.x + threadIdx.x;
    if (i < n) deg[i] = 1.0f;  // self-loop contributes 1 to every node
}

__global__ void deg_edge_kernel(float* deg, const long long* __restrict__ dst, int e) {
    int i = blockIdx.x * blockDim.x + threadIdx.x;
    if (i < e) atomicAdd(&deg[(int)dst[i]], 1.0f);
}

__global__ void deg_rsqrt_kernel(float* deg, int n) {
    int i = blockIdx.x * blockDim.x + threadIdx.x;
    if (i < n) deg[i] = rsqrtf(deg[i]);  // deg >= 1 always (self-loops)
}

// ---------------- GEMM: h = x @ W^T via V_WMMA_F32_16X16X4_F32 ----------------
// One wave (32 lanes) computes a 16-row x 128-col panel of h.
// f32 A-matrix 16x4 layout: lanes 0-15 -> M=lane; VGPR0 = K=0 (lo half)/K=2 (hi half),
// VGPR1 = K=1/K=3.  B (4x16 = W^T tile) mirrored: lane -> N, halves -> K pairs.
__global__ __launch_bounds__(128) void gemm_xwT_kernel(const float* __restrict__ x,
                                                       const float* __restrict__ W,
                                                       float* __restrict__ h, int n) {
    const int wave = threadIdx.x >> 5;
    const int lane = threadIdx.x & 31;
    const int half = lane >> 4;     // 0: lanes 0-15, 1: lanes 16-31
    const int r    = lane & 15;
    const int m0   = (blockIdx.x * 4 + wave) * 16;
    if (m0 + 16 > n) return;        // whole wave exits together (EXEC all-ones for WMMA)

    v8f acc[8] = {};                // 8 N-tiles of 16 -> full 128 output cols

    const float* xrow = x + (size_t)(m0 + r) * D + 2 * half;
#pragma unroll 4
    for (int k0 = 0; k0 < D; k0 += 4) {
        v2f a = *(const v2f*)(xrow + k0);
#pragma unroll
        for (int t = 0; t < 8; ++t) {
            // B[k][nn] = W^T[k0+k][16t+nn] = W[16t+nn][k0+k]
            v2f b = *(const v2f*)(W + (size_t)(t * 16 + r) * D + k0 + 2 * half);
            acc[t] = __builtin_amdgcn_wmma_f32_16x16x4_f32(
                false, a, false, b, (short)0, acc[t], false, false);
        }
    }

    // C/D layout: VGPR v -> M = v (lanes 0-15) / v+8 (lanes 16-31), N = lane%16
#pragma unroll
    for (int t = 0; t < 8; ++t) {
        float* hp = h + (size_t)(m0 + 8 * half) * D + t * 16 + r;
#pragma unroll
        for (int v = 0; v < 8; ++v) {
            hp[(size_t)v * D] = acc[t][v];
        }
    }
}

// ---------------- out = bias + dinv[n]^2 * h[n]  (self-loop term) ----------------

__global__ void init_out_kernel(float* __restrict__ out, const float* __restrict__ h,
                                const float* __restrict__ dinv,
                                const float* __restrict__ bias, int n) {
    int i = blockIdx.x * blockDim.x + threadIdx.x;   // one thread per (node, col/4)
    int node = i >> 5;
    int c = (i & 31) * 4;
    if (node >= n) return;
    float dn = dinv[node];
    float s = dn * dn;
    float4 hv = *(const float4*)(h + (size_t)node * D + c);
    float4 bv = *(const float4*)(bias + c);
    float4 o;
    o.x = bv.x + s * hv.x;
    o.y = bv.y + s * hv.y;
    o.z = bv.z + s * hv.z;
    o.w = bv.w + s * hv.w;
    *(float4*)(out + (size_t)node * D + c) = o;
}

// ---------------- edge scatter: out[dst] += dinv[src]*dinv[dst] * h[src] ----------------
// One wave per edge; each lane owns a float4 slice of the 128-wide feature row.
__global__ __launch_bounds__(256) void scatter_kernel(float* __restrict__ out,
                                                      const float* __restrict__ h,
                                                      const float* __restrict__ dinv,
                                                      const long long* __restrict__ src,
                                                      const long long* __restrict__ dst,
                                                      int e) {
    const int wave = threadIdx.x >> 5;
    const int lane = threadIdx.x & 31;
    const int eid = blockIdx.x * 8 + wave;
    if (eid >= e) return;
    const int s = (int)src[eid];
    const int d = (int)dst[eid];
    const float norm = dinv[s] * dinv[d];
    const float4 hv = *(const float4*)(h + (size_t)s * D + lane * 4);
    float* op = out + (size_t)d * D + lane * 4;
    atomicAdd(op + 0, norm * hv.x);
    atomicAdd(op + 1, norm * hv.y);
    atomicAdd(op + 2, norm * hv.z);
    atomicAdd(op + 3, norm * hv.w);
}

// ---------------- driver ----------------

extern "C" void kernel_launch(void* const* d_in, const int* in_sizes, int n_in,
                              void* d_out, int out_size, void* d_ws, size_t ws_size,
                              hipStream_t stream) {
    // setup_inputs order: x_r1, x_r2, x_i1, x_i2, edge_index, W1, b1, W2, b2
    const float* x[4] = {(const float*)d_in[0], (const float*)d_in[1],
                         (const float*)d_in[2], (const float*)d_in[3]};
    const long long* ei = (const long long*)d_in[4];   // int64 [2, E]
    const float* W1 = (const float*)d_in[5];
    const float* b1 = (const float*)d_in[6];
    const float* W2 = (const float*)d_in[7];
    const float* b2 = (const float*)d_in[8];

    const int N = in_sizes[0] / D;     // 50000
    const int E = in_sizes[4] / 2;     // 800000
    const long long* srcp = ei;        // row 0
    const long long* dstp = ei + E;    // row 1

    float* dinv = (float*)d_ws;                 // N floats (deg -> rsqrt in place)
    float* hbuf = dinv + 65536;                 // N*D floats, reused per matrix

    // normalization coefficients (shared by all 4 convs)
    deg_init_kernel<<<(N + 255) / 256, 256, 0, stream>>>(dinv, N);
    deg_edge_kernel<<<(E + 255) / 256, 256, 0, stream>>>(dinv, dstp, E);
    deg_rsqrt_kernel<<<(N + 255) / 256, 256, 0, stream>>>(dinv, N);

    const float* Ws[4] = {W1, W2, W1, W2};
    const float* bs[4] = {b1, b2, b1, b2};
    float* outp = (float*)d_out;

    const int mtiles = N / 16;                       // 3125
    const int gblocks = (mtiles + 3) / 4;            // 4 waves/block

    for (int t = 0; t < 4; ++t) {
        float* out_t = outp + (size_t)t * N * D;
        gemm_xwT_kernel<<<gblocks, 128, 0, stream>>>(x[t], Ws[t], hbuf, N);
        init_out_kernel<<<(N * 32 + 255) / 256, 256, 0, stream>>>(out_t, hbuf, dinv, bs[t], N);
        scatter_kernel<<<(E + 7) / 8, 256, 0, stream>>>(out_t, hbuf, dinv, srcp, dstp, E);
    }
}